// SymmetricThicknessLoss_31516470018694
// MI455X (gfx1250) — compile-verified
//
#include <hip/hip_runtime.h>
#include <hip/hip_bf16.h>

// ---------------------------------------------------------------------------
// SymmetricThicknessLoss for B=4, N=8192 point clouds.
// Dominant cost: 4 brute-force NN-argmin passes (1.07G distance evals).
// V_WMMA_F32_16X16X4_F32 computes a full 16x16 tile of
//   d[s][q] = ||s||^2 - 2 s.q
// per instruction via K=3 -> K=4 augmentation:
//   A[s] = (-2sx, -2sy, -2sz, ||s||^2),  B[:,q] = (qx, qy, qz, 1)
// Argmin: tournament tree per 8-value WMMA output (depth-4 dependency chain
// instead of a serial depth-8 running-min scan), halves merged via
// shfl_xor(16) with exact first-occurrence tie-breaking.
// ---------------------------------------------------------------------------

typedef __attribute__((ext_vector_type(2))) float v2f;
typedef __attribute__((ext_vector_type(8))) float v8f;

#define BSZ   4
#define NPTS  8192
#define CHUNK 2048                 // search points staged in LDS (32 KB as float4)
#define WAVES_PER_BLOCK 8
#define QUERIES_PER_BLOCK (WAVES_PER_BLOCK * 16)

// Tournament node: prefer 'a' (lower index) on ties -> first occurrence.
__device__ __forceinline__ void argmin_node(float va, int ia, float vb, int ib,
                                            float& vo, int& io)
{
    const bool t = vb < va;
    vo = t ? vb : va;
    io = t ? ib : ia;
}

// ---------------------------------------------------------------------------
// Kernel 1: brute-force NN argmin via WMMA.
// grid = (NPTS/QUERIES_PER_BLOCK, BSZ, 4 tasks), block = 256 (8 waves).
// task 0: q=pred_white, s=pred_pial   -> i1 (pred)
// task 1: q=pred_pial,  s=pred_white  -> i2 (pred)
// task 2: q=true_white, s=true_pial   -> i1 (true)
// task 3: q=true_pial,  s=true_white  -> i2 (true)
// ---------------------------------------------------------------------------
__global__ __launch_bounds__(256) void nn_argmin_wmma_kernel(
    const float* __restrict__ pw, const float* __restrict__ pp,
    const float* __restrict__ tw, const float* __restrict__ tp,
    int* __restrict__ nn_out /* [4][BSZ][NPTS] */)
{
    __shared__ float4 sPts[CHUNK];   // (-2x, -2y, -2z, ||s||^2)

    const int task = blockIdx.z;
    const int b    = blockIdx.y;

    const float* q = pw;
    const float* s = pp;
    if (task == 1) { q = pp; s = pw; }
    else if (task == 2) { q = tw; s = tp; }
    else if (task == 3) { q = tp; s = tw; }
    q += (size_t)b * NPTS * 3;
    s += (size_t)b * NPTS * 3;
    int* out = nn_out + ((size_t)task * BSZ + b) * NPTS;

    const int lane = threadIdx.x & 31;
    const int wave = threadIdx.x >> 5;
    const int col  = lane & 15;                       // query column in tile
    const bool lo  = (lane < 16);
    const int qIdx = blockIdx.x * QUERIES_PER_BLOCK + wave * 16 + col;

    // B fragment (query matrix, 4x16): register-resident for the whole kernel.
    // lanes 0-15 : (B[0][col], B[1][col]) = (qx, qy)
    // lanes 16-31: (B[2][col], B[3][col]) = (qz, 1)
    const float qx = q[qIdx * 3 + 0];
    const float qy = q[qIdx * 3 + 1];
    const float qz = q[qIdx * 3 + 2];
    v2f bfrag;
    bfrag[0] = lo ? qx : qz;
    bfrag[1] = lo ? qy : 1.0f;

    float minv = 3.0e38f;
    int   mini = 0;

    for (int chunk = 0; chunk < NPTS; chunk += CHUNK) {
        __syncthreads();
        // Stage CHUNK augmented search points into LDS cooperatively.
        for (int i = threadIdx.x; i < CHUNK; i += 256) {
            const int si = chunk + i;
            const float sx = s[si * 3 + 0];
            const float sy = s[si * 3 + 1];
            const float sz = s[si * 3 + 2];
            sPts[i] = make_float4(-2.0f * sx, -2.0f * sy, -2.0f * sz,
                                  sx * sx + sy * sy + sz * sz);
        }
        __syncthreads();

#pragma unroll 2
        for (int t = 0; t < CHUNK; t += 16) {
            // A fragment (16x4, one search point per row):
            // lanes 0-15 : (A[row][0], A[row][1]), row = col
            // lanes 16-31: (A[row][2], A[row][3]), row = col
            const float* sp = ((const float*)&sPts[t + col]) + (lo ? 0 : 2);
            v2f afrag;
            afrag[0] = sp[0];
            afrag[1] = sp[1];

            v8f c = {};
            c = __builtin_amdgcn_wmma_f32_16x16x4_f32(
                    /*neg_a=*/false, afrag, /*neg_b=*/false, bfrag,
                    /*c_mod=*/(short)0, c, /*reuse_a=*/false, /*reuse_b=*/false);

            // Rows base..base+7 in this lane; indices strictly increasing, so
            // a left-preferring tournament keeps the first occurrence.
            const int base = chunk + t + (lo ? 0 : 8);

            float v01, v23, v45, v67, v03, v47, vt;
            int   i01, i23, i45, i67, i03, i47, it;
            argmin_node(c[0], base + 0, c[1], base + 1, v01, i01);
            argmin_node(c[2], base + 2, c[3], base + 3, v23, i23);
            argmin_node(c[4], base + 4, c[5], base + 5, v45, i45);
            argmin_node(c[6], base + 6, c[7], base + 7, v67, i67);
            argmin_node(v01, i01, v23, i23, v03, i03);
            argmin_node(v45, i45, v67, i67, v47, i47);
            argmin_node(v03, i03, v47, i47, vt, it);
            // Running min holds older (smaller) indices: keep it on ties.
            argmin_node(minv, mini, vt, it, minv, mini);
        }
    }

    // Merge lane L (rows 0-7 of each tile) with lane L+16 (rows 8-15),
    // exact first-occurrence tie-break on equal distance.
    const float ov = __shfl_xor(minv, 16);
    const int   oi = __shfl_xor(mini, 16);
    const bool take = (ov < minv) || (ov == minv && oi < mini);
    minv = take ? ov : minv;
    mini = take ? oi : mini;

    if (lo) out[qIdx] = mini;
}

// ---------------------------------------------------------------------------
// Kernel 2: thickness vectors.
//   yp_inner[b,n] = pp[b, i1p[b,n]] - pw[b,n]
//   yp_outer[b,n] = pp[b,n] - pw[b, i2p[b,n]]
//   yt_inner[b,n] = tp[b, i1t[b,n]] - tw[b,n]
//   yt_outer[b,n] = tp[b,n] - tw[b, i2t[b,n]]
// th layout: [4][BSZ*NPTS*3] : 0=yp_inner, 1=yp_outer, 2=yt_inner, 3=yt_outer
// ---------------------------------------------------------------------------
__global__ __launch_bounds__(256) void thickness_kernel(
    const float* __restrict__ pw, const float* __restrict__ pp,
    const float* __restrict__ tw, const float* __restrict__ tp,
    const int* __restrict__ nn, float* __restrict__ th)
{
    const int gid = blockIdx.x * blockDim.x + threadIdx.x;
    if (gid >= BSZ * NPTS) return;
    const int b = gid / NPTS;
    const int n = gid - b * NPTS;

    const int i1p = nn[(0 * BSZ + b) * NPTS + n];
    const int i2p = nn[(1 * BSZ + b) * NPTS + n];
    const int i1t = nn[(2 * BSZ + b) * NPTS + n];
    const int i2t = nn[(3 * BSZ + b) * NPTS + n];

    const size_t row = (size_t)(b * NPTS + n) * 3;
    const size_t rbase = (size_t)b * NPTS * 3;
    const size_t stride = (size_t)BSZ * NPTS * 3;

#pragma unroll
    for (int d = 0; d < 3; ++d) {
        th[0 * stride + row + d] = pp[rbase + (size_t)i1p * 3 + d] - pw[row + d];
        th[1 * stride + row + d] = pp[row + d] - pw[rbase + (size_t)i2p * 3 + d];
        th[2 * stride + row + d] = tp[rbase + (size_t)i1t * 3 + d] - tw[row + d];
        th[3 * stride + row + d] = tp[row + d] - tw[rbase + (size_t)i2t * 3 + d];
    }
}

// ---------------------------------------------------------------------------
// Kernel 3: fused symmetric MSN reduction into a scalar.
// loss = 0.25/(B*N) * sum_{b,n} (termA + termB + termC + termD)
// ---------------------------------------------------------------------------
__global__ void zero_out_kernel(float* out) { *out = 0.0f; }

__device__ inline float sqdiff3(const float* __restrict__ a, size_t ia,
                                const float* __restrict__ b, size_t ib)
{
    const float d0 = a[ia + 0] - b[ib + 0];
    const float d1 = a[ia + 1] - b[ib + 1];
    const float d2 = a[ia + 2] - b[ib + 2];
    return d0 * d0 + d1 * d1 + d2 * d2;
}

__global__ __launch_bounds__(256) void loss_reduce_kernel(
    const float* __restrict__ th,
    const int* __restrict__ i_pw, const int* __restrict__ i_tw,
    const int* __restrict__ i_pp, const int* __restrict__ i_tp,
    float* __restrict__ out)
{
    __shared__ float warpSums[WAVES_PER_BLOCK];

    const int gid = blockIdx.x * blockDim.x + threadIdx.x;
    float acc = 0.0f;
    if (gid < BSZ * NPTS) {
        const int b = gid / NPTS;
        const int n = gid - b * NPTS;

        const size_t stride = (size_t)BSZ * NPTS * 3;
        const float* yp_in = th + 0 * stride;
        const float* yp_ou = th + 1 * stride;
        const float* yt_in = th + 2 * stride;
        const float* yt_ou = th + 3 * stride;

        const size_t row   = (size_t)(b * NPTS + n) * 3;
        const size_t rbase = (size_t)b * NPTS * 3;

        const size_t gA = rbase + (size_t)i_pw[b * NPTS + n] * 3;
        const size_t gB = rbase + (size_t)i_tw[b * NPTS + n] * 3;
        const size_t gC = rbase + (size_t)i_pp[b * NPTS + n] * 3;
        const size_t gD = rbase + (size_t)i_tp[b * NPTS + n] * 3;

        acc  = sqdiff3(yp_in, row, yt_in, gA);   // msn(yp_inner, yt_inner[i_pred_white])
        acc += sqdiff3(yp_in, gB,  yt_in, row);  // msn(yp_inner[i_true_white], yt_inner)
        acc += sqdiff3(yp_ou, row, yt_ou, gC);   // msn(yp_outer, yt_outer[i_pred_pial])
        acc += sqdiff3(yp_ou, gD,  yt_ou, row);  // msn(yp_outer[i_true_pial], yt_outer)
    }

    // wave32 reduce
#pragma unroll
    for (int off = 16; off > 0; off >>= 1)
        acc += __shfl_down(acc, off);

    const int lane = threadIdx.x & 31;
    const int wave = threadIdx.x >> 5;
    if (lane == 0) warpSums[wave] = acc;
    __syncthreads();

    if (wave == 0) {
        float v = (lane < WAVES_PER_BLOCK) ? warpSums[lane] : 0.0f;
#pragma unroll
        for (int off = 16; off > 0; off >>= 1)
            v += __shfl_down(v, off);
        if (lane == 0)
            atomicAdd(out, v * (0.25f / (float)(BSZ * NPTS)));
    }
}

// ---------------------------------------------------------------------------
// Launcher
// ---------------------------------------------------------------------------
extern "C" void kernel_launch(void* const* d_in, const int* in_sizes, int n_in,
                              void* d_out, int out_size, void* d_ws, size_t ws_size,
                              hipStream_t stream)
{
    const float* pw = (const float*)d_in[0];   // pred_white [B,N,3]
    const float* pp = (const float*)d_in[1];   // pred_pial
    const float* tw = (const float*)d_in[2];   // true_white
    const float* tp = (const float*)d_in[3];   // true_pial
    const int* i_pw = (const int*)d_in[4];     // i_pred_white [B,N]
    const int* i_tw = (const int*)d_in[5];     // i_true_white
    const int* i_pp = (const int*)d_in[6];     // i_pred_pial
    const int* i_tp = (const int*)d_in[7];     // i_true_pial

    // Workspace: [4][B][N] int32 NN indices, then [4][B*N*3] f32 thickness.
    int*   nn = (int*)d_ws;
    float* th = (float*)((char*)d_ws + (size_t)4 * BSZ * NPTS * sizeof(int));

    float* out = (float*)d_out;

    zero_out_kernel<<<1, 1, 0, stream>>>(out);

    dim3 g1(NPTS / QUERIES_PER_BLOCK, BSZ, 4);
    nn_argmin_wmma_kernel<<<g1, 256, 0, stream>>>(pw, pp, tw, tp, nn);

    const int total = BSZ * NPTS;
    thickness_kernel<<<(total + 255) / 256, 256, 0, stream>>>(pw, pp, tw, tp, nn, th);

    loss_reduce_kernel<<<(total + 255) / 256, 256, 0, stream>>>(
        th, i_pw, i_tw, i_pp, i_tp, out);
}